// LocalEntropy_49289044688944
// MI455X (gfx1250) — compile-verified
//
#include <hip/hip_runtime.h>
#include <math.h>

#ifndef __has_builtin
#define __has_builtin(x) 0
#endif

#define IMG      512
#define TILE_W   64
#define TILE_H   32
#define HALO     2
#define IN_W     (TILE_W + 2 * HALO)   /* 68 */
#define IN_H     (TILE_H + 2 * HALO)   /* 36 */
#define NTHREADS 256

#define HALF_LOG_2PIE 1.4189385332046727f  /* 0.5*ln(2*pi*e) */
#define EPS_F 1e-6f

typedef __attribute__((address_space(1))) int g1_int;
typedef __attribute__((address_space(3))) int l3_int;

// ---- CDNA5 async global->LDS copy (ASYNCcnt-tracked, bypasses VGPRs) ----
__device__ __forceinline__ void async_ld_b32(const float* g, float* l) {
#if __has_builtin(__builtin_amdgcn_global_load_async_to_lds_b32)
  __builtin_amdgcn_global_load_async_to_lds_b32(
      (g1_int*)(__UINTPTR_TYPE__)g,
      (l3_int*)(unsigned)(__UINTPTR_TYPE__)l,
      0, 0);
#else
  unsigned           lo = (unsigned)(__UINTPTR_TYPE__)l;
  unsigned long long ga = (unsigned long long)(__UINTPTR_TYPE__)g;
  asm volatile("global_load_async_to_lds_b32 %0, %1, off"
               :: "v"(lo), "v"(ga) : "memory");
#endif
}

__device__ __forceinline__ void wait_async_0() {
#if __has_builtin(__builtin_amdgcn_s_wait_asynccnt)
  __builtin_amdgcn_s_wait_asynccnt(0);
#else
  asm volatile("s_wait_asynccnt 0" ::: "memory");
#endif
}

__global__ __launch_bounds__(NTHREADS)
void LocalEntropy_gfx1250_kernel(const float* __restrict__ x,
                                 float* __restrict__ out) {
  // alignas(16): phase 2 issues ds_load_b128 / merged b128 stores; LDS B128
  // access requires 16-byte natural alignment (DWORD mode would silently
  // mask low address bits otherwise).
  __shared__ alignas(16) float  xs[IN_H * IN_W];   // raw input tile (zero-padded halo)
  __shared__ alignas(16) float2 rs[IN_H * TILE_W]; // per-row (sum, sum_sq), 5-wide windows

  const int  tid   = threadIdx.x;
  const int  tx0   = blockIdx.x * TILE_W;
  const int  ty0   = blockIdx.y * TILE_H;
  const long plane = blockIdx.z;                  // b*64 + c
  const float* __restrict__ gp = x   + plane * (long)(IMG * IMG);
  float*       __restrict__ op = out + plane * (long)(IMG * IMG);

  // ---------------- Phase 1: async stage input tile into LDS ----------------
  for (int idx = tid; idx < IN_H * IN_W; idx += NTHREADS) {
    int r  = idx / IN_W;
    int c  = idx - r * IN_W;
    int gy = ty0 - HALO + r;
    int gx = tx0 - HALO + c;
    if ((unsigned)gy < (unsigned)IMG && (unsigned)gx < (unsigned)IMG) {
      async_ld_b32(&gp[gy * IMG + gx], &xs[idx]);
    } else {
      xs[idx] = 0.0f;  // zero padding (matches Unfold SAME padding)
    }
  }
  wait_async_0();
  __syncthreads();

  // ------- Phase 2: horizontal 5-tap rolling sums of x and x^2 -------------
  // Each segment = one row r, 8 consecutive outputs; rolling window in regs.
  for (int seg = tid; seg < IN_H * (TILE_W / 8); seg += NTHREADS) {
    int r  = seg >> 3;
    int j0 = (seg & 7) * 8;
    const float4* rv = reinterpret_cast<const float4*>(&xs[r * IN_W + j0]);
    float4 a = rv[0], b = rv[1], c = rv[2];
    float v[12] = {a.x, a.y, a.z, a.w, b.x, b.y, b.z, b.w, c.x, c.y, c.z, c.w};
    float q[12];
#pragma unroll
    for (int k = 0; k < 12; ++k) q[k] = v[k] * v[k];
    float s  = v[0] + v[1] + v[2] + v[3] + v[4];
    float ss = q[0] + q[1] + q[2] + q[3] + q[4];
    float2* orow = &rs[r * TILE_W + j0];
    orow[0] = make_float2(s, ss);
#pragma unroll
    for (int k = 1; k < 8; ++k) {
      s  += v[k + 4] - v[k - 1];
      ss += q[k + 4] - q[k - 1];
      orow[k] = make_float2(s, ss);
    }
  }
  __syncthreads();

  // ------- Phase 3: vertical 5-tap rolling sum + entropy + NT store --------
  const int j  = tid & (TILE_W - 1);     // column within tile (coalesced)
  const int y0 = (tid >> 6) * 8;         // 8-row vertical strip per thread
  float2 h[12];
#pragma unroll
  for (int k = 0; k < 12; ++k) h[k] = rs[(y0 + k) * TILE_W + j];
  float s  = h[0].x + h[1].x + h[2].x + h[3].x + h[4].x;
  float ss = h[0].y + h[1].y + h[2].y + h[3].y + h[4].y;
  float* ocol = &op[(long)(ty0 + y0) * IMG + tx0 + j];
#pragma unroll
  for (int k = 0; k < 8; ++k) {
    // var = (ss - s^2/25)/24 + eps ; out = 0.5*ln(2*pi*e*var)
    float var = (ss - s * s * (1.0f / 25.0f)) * (1.0f / 24.0f) + EPS_F;
    float res = 0.5f * __logf(var) + HALF_LOG_2PIE;
    __builtin_nontemporal_store(res, ocol + (long)k * IMG);  // streamed output
    if (k < 7) {
      s  += h[k + 5].x - h[k].x;
      ss += h[k + 5].y - h[k].y;
    }
  }
}

extern "C" void kernel_launch(void* const* d_in, const int* in_sizes, int n_in,
                              void* d_out, int out_size, void* d_ws, size_t ws_size,
                              hipStream_t stream) {
  (void)in_sizes; (void)n_in; (void)d_ws; (void)ws_size; (void)out_size;
  const float* x  = (const float*)d_in[0];
  float*      out = (float*)d_out;
  dim3 grid(IMG / TILE_W, IMG / TILE_H, 8 * 64);  // 8 x 16 x 512 tiles
  dim3 block(NTHREADS);
  hipLaunchKernelGGL(LocalEntropy_gfx1250_kernel, grid, block, 0, stream, x, out);
}